// GLCN_49117245997723
// MI455X (gfx1250) — compile-verified
//
#include <hip/hip_runtime.h>
#include <math.h>

// Problem constants (from reference): B=4, N=1024, F=128, FOBS=32, D=128
#define BATCH 4
#define NPTS  1024
#define FF    128
#define NOBS  32
#define DD    128

typedef float v2f __attribute__((ext_vector_type(2)));
typedef float v8f __attribute__((ext_vector_type(8)));

// ---------------------------------------------------------------------------
// Kernel 0: zero the Ds output region, copy X -> X output region, and seed
// the per-row degree sums with 1.0 (the +I diagonal of A_hat).
// ---------------------------------------------------------------------------
__global__ void glcn_init_kernel(const float* __restrict__ Xin,
                                 float* __restrict__ out,
                                 float* __restrict__ dsum)
{
    const long idx = (long)blockIdx.x * blockDim.x + threadIdx.x;
    const long HS  = (long)BATCH * NPTS * DD;       // Hs elements
    const long AS  = (long)BATCH * NPTS * NPTS;     // As elements
    const long XO  = HS + AS;                       // X passthrough offset
    const long DO  = XO + (long)BATCH * NPTS * FF;  // Ds offset
    if (idx < AS)                     out[DO + idx] = 0.0f;     // Ds = 0
    if (idx < (long)BATCH*NPTS*FF)    out[XO + idx] = Xin[idx]; // X copy
    if (idx < (long)BATCH*NPTS)       dsum[idx]     = 1.0f;     // diag seed
}

// ---------------------------------------------------------------------------
// Kernel 1: adjacency. 32x32 tile of A per block (256 threads).
// e[i,j] = sum_f |h_i[f]-h_j[f]| * a[f]  (f < 32), A = sigmoid(e), zero diag.
// Also accumulates row sums of A into dsum via LDS + global atomics.
// ---------------------------------------------------------------------------
__global__ void glcn_adj_kernel(const float* __restrict__ X,
                                const float* __restrict__ a_link,
                                float* __restrict__ Aout,
                                float* __restrict__ dsum)
{
    __shared__ float hi[32 * 32];
    __shared__ float hj[32 * 32];
    __shared__ float as[32];
    __shared__ float rs[32];

    const int b  = blockIdx.z;
    const int i0 = blockIdx.y * 32;
    const int j0 = blockIdx.x * 32;
    const int t  = threadIdx.x;

    {   // cooperative load of the two 32x32 h tiles (first NOBS features)
        const int row = t >> 3;           // 0..31
        const int f4  = (t & 7) * 4;      // 0,4,...,28
        *(float4*)&hi[row * 32 + f4] =
            *(const float4*)&X[((long)b * NPTS + i0 + row) * FF + f4];
        *(float4*)&hj[row * 32 + f4] =
            *(const float4*)&X[((long)b * NPTS + j0 + row) * FF + f4];
        if (t < 32) { as[t] = a_link[t]; rs[t] = 0.0f; }
    }
    __syncthreads();

    const int jl = t & 31;                // column within tile (coalesced)
    const int ib = t >> 5;                // 0..7
    #pragma unroll
    for (int ii = 0; ii < 4; ++ii) {
        const int il = ib + ii * 8;       // row within tile
        float s = 0.0f;
        #pragma unroll 8
        for (int f = 0; f < NOBS; ++f)
            s += fabsf(hi[il * 32 + f] - hj[jl * 32 + f]) * as[f];
        float a = 1.0f / (1.0f + __expf(-s));
        if (i0 + il == j0 + jl) a = 0.0f;
        Aout[(long)b * NPTS * NPTS + (long)(i0 + il) * NPTS + (j0 + jl)] = a;
        atomicAdd(&rs[il], a);
    }
    __syncthreads();
    if (t < 32) atomicAdd(&dsum[b * NPTS + i0 + t], rs[t]);
}

// ---------------------------------------------------------------------------
// Kernel 2: dsum -> d^{-1/2}
// ---------------------------------------------------------------------------
__global__ void glcn_dinv_kernel(float* __restrict__ dsum)
{
    const int i = blockIdx.x * blockDim.x + threadIdx.x;
    if (i < BATCH * NPTS) dsum[i] = rsqrtf(dsum[i]);
}

// ---------------------------------------------------------------------------
// Kernel 3: norm[b,i,j] = dinv[i] * (A[b,i,j] + (i==j)) * dinv[j]
// ---------------------------------------------------------------------------
__global__ void glcn_norm_kernel(const float* __restrict__ Ain,
                                 const float* __restrict__ dinv,
                                 float* __restrict__ normw)
{
    const long idx = (long)blockIdx.x * blockDim.x + threadIdx.x; // < B*N*N
    const int  j   = (int)(idx & (NPTS - 1));
    const long bi  = idx >> 10;
    const int  i   = (int)(bi & (NPTS - 1));
    const int  b   = (int)(bi >> 10);
    float a = Ain[idx] + ((i == j) ? 1.0f : 0.0f);
    normw[idx] = dinv[b * NPTS + i] * a * dinv[b * NPTS + j];
}

// ---------------------------------------------------------------------------
// Kernel 4: fp32 WMMA GEMM  C[M x 128] = A[M x K] @ B[K x 128]  (M = 1024)
// Block = 256 threads = 8 waves; block computes a 16-row x 128-col strip.
// The 16 x 64 A tile is staged in LDS and shared by all 8 waves (8x reuse of
// the big `norm` operand). Each wave runs a 16x16x4 v_wmma_f32 K-loop.
//
// VGPR layouts per CDNA5 ISA 7.12.2 (g = lane/16, m/n = lane%16):
//   A frag: VGPR r holds A[m][k0 + 2g + r]  -> consecutive float2 per lane
//   B frag: VGPR r holds B[k0 + 2g + r][n]
//   C/D   : VGPR r holds C[r + 8g][n]
// ---------------------------------------------------------------------------
#define KCHUNK 64
__global__ void glcn_gemm_wmma(const float* __restrict__ A, long a_bstride, int lda,
                               const float* __restrict__ Bm, long b_bstride,
                               float* __restrict__ C, long c_bstride,
                               int K, int apply_relu)
{
    __shared__ float atile[16 * KCHUNK];

    const int b    = blockIdx.y;
    const int i0   = blockIdx.x * 16;
    const int t    = threadIdx.x;
    const int lane = t & 31;
    const int wave = t >> 5;          // 0..7
    const int n0   = wave * 16;
    const int mn   = lane & 15;       // M row (A frag) / N col (B,C frags)
    const int g    = lane >> 4;       // lane-group

    const float* Ab = A  + (long)b * a_bstride;
    const float* Bb = Bm + (long)b * b_bstride;

    v8f acc = {0.f, 0.f, 0.f, 0.f, 0.f, 0.f, 0.f, 0.f};

    const int row_ld = t >> 4;        // 0..15
    const int col_ld = (t & 15) * 4;  // 0..60

    for (int kc = 0; kc < K; kc += KCHUNK) {
        // cooperative stage of 16 x KCHUNK A tile (contiguous float4 loads)
        *(float4*)&atile[row_ld * KCHUNK + col_ld] =
            *(const float4*)&Ab[(long)(i0 + row_ld) * lda + kc + col_ld];
        __syncthreads();

        #pragma unroll
        for (int ks = 0; ks < KCHUNK / 4; ++ks) {
            const int kk = ks * 4 + 2 * g;
            v2f afrag;
            afrag.x = atile[mn * KCHUNK + kk];
            afrag.y = atile[mn * KCHUNK + kk + 1];
            v2f bfrag;
            bfrag.x = Bb[(long)(kc + kk)     * DD + n0 + mn];
            bfrag.y = Bb[(long)(kc + kk + 1) * DD + n0 + mn];
            acc = __builtin_amdgcn_wmma_f32_16x16x4_f32(
                      false, afrag, false, bfrag,
                      (short)0, acc, false, false);
        }
        __syncthreads();
    }

    float* Cb = C + (long)b * c_bstride;
    #pragma unroll
    for (int r = 0; r < 8; ++r) {
        float v = acc[r];
        if (apply_relu) v = fmaxf(v, 0.0f);
        Cb[(long)(i0 + r + 8 * g) * DD + n0 + mn] = v;
    }
}

// ---------------------------------------------------------------------------
// Host launcher
// ---------------------------------------------------------------------------
extern "C" void kernel_launch(void* const* d_in, const int* in_sizes, int n_in,
                              void* d_out, int out_size, void* d_ws, size_t ws_size,
                              hipStream_t stream)
{
    (void)in_sizes; (void)n_in; (void)out_size; (void)ws_size;

    const float* X      = (const float*)d_in[0];   // (B,N,F)
    const float* a_link = (const float*)d_in[1];   // (FOBS,1)
    const float* W0     = (const float*)d_in[2];   // (F,D)
    const float* W1     = (const float*)d_in[3];   // (D,D)
    const float* W2     = (const float*)d_in[4];   // (D,D)

    float* out = (float*)d_out;
    float* ws  = (float*)d_ws;

    // workspace layout (floats)
    float* normw = ws;                                 // B*N*N   (16 MB)
    float* buf0  = normw + (long)BATCH * NPTS * NPTS;  // B*N*D
    float* buf1  = buf0  + (long)BATCH * NPTS * DD;    // B*N*D
    float* dsum  = buf1  + (long)BATCH * NPTS * DD;    // B*N

    // output layout (floats): Hs | As | X | Ds
    float* Hs = out;
    float* As = out + (long)BATCH * NPTS * DD;

    const long AS_ELEMS = (long)BATCH * NPTS * NPTS;   // 4194304

    // 0) init: zero Ds, copy X through, seed dsum with diagonal
    glcn_init_kernel<<<(int)(AS_ELEMS / 256), 256, 0, stream>>>(X, out, dsum);

    // 1) adjacency A = sigmoid(e), zero diag; accumulate row sums
    glcn_adj_kernel<<<dim3(NPTS / 32, NPTS / 32, BATCH), 256, 0, stream>>>(
        X, a_link, As, dsum);

    // 2) dsum -> d^{-1/2}
    glcn_dinv_kernel<<<(BATCH * NPTS) / 256, 256, 0, stream>>>(dsum);

    // 3) norm = D^{-1/2} (A + I) D^{-1/2}
    glcn_norm_kernel<<<(int)(AS_ELEMS / 256), 256, 0, stream>>>(As, dsum, normw);

    const dim3 ggrid(NPTS / 16, BATCH);
    const long sNN = (long)NPTS * NPTS;
    const long sND = (long)NPTS * DD;
    const long sNF = (long)NPTS * FF;

    // 4) T0 = X @ W0
    glcn_gemm_wmma<<<ggrid, 256, 0, stream>>>(X, sNF, FF,  W0, 0,   buf0, sND, FF,   0);
    // 5) H1 = norm @ T0
    glcn_gemm_wmma<<<ggrid, 256, 0, stream>>>(normw, sNN, NPTS, buf0, sND, buf1, sND, NPTS, 0);
    // 6) T1 = H1 @ W1
    glcn_gemm_wmma<<<ggrid, 256, 0, stream>>>(buf1, sND, DD,  W1, 0,   buf0, sND, DD,   0);
    // 7) H2 = norm @ T1
    glcn_gemm_wmma<<<ggrid, 256, 0, stream>>>(normw, sNN, NPTS, buf0, sND, buf1, sND, NPTS, 0);
    // 8) T2 = H2 @ W2
    glcn_gemm_wmma<<<ggrid, 256, 0, stream>>>(buf1, sND, DD,  W2, 0,   buf0, sND, DD,   0);
    // 9) Hs = relu(norm @ T2)
    glcn_gemm_wmma<<<ggrid, 256, 0, stream>>>(normw, sNN, NPTS, buf0, sND, Hs, sND, NPTS, 1);
}